// DCTNLAttention11_63848983822660
// MI455X (gfx1250) — compile-verified
//
#include <hip/hip_runtime.h>
#include <math.h>

// ---------------- problem constants ----------------
constexpr int Bb  = 8;
constexpr int Cc  = 512;
constexpr int Hh  = 128;
constexpr int Wd  = 128;
constexpr int HWn = Hh * Wd;     // 16384
constexpr int Kk  = 64;          // DCT band 8x8
constexpr int ODn = 64;
constexpr int NCH = 32;          // split-K chunks for the A accumulation

// ---------------- workspace layout (floats) ----------------
constexpr size_t F_Dh = 0;                                   // 128*8
constexpr size_t F_Dv = F_Dh + 1024;                         // 128*8
constexpr size_t F_sP = F_Dv + 1024;                         // 64 (column sums of P)
constexpr size_t F_PT = 4096;                                // 64*HW  (P^T, k-major)
constexpr size_t F_xP = F_PT + (size_t)Kk * HWn;             // B*C*64
constexpr size_t F_Wq = F_xP + (size_t)Bb * Cc * Kk;         // B*64*64  (WqxP)
constexpr size_t F_Wk = F_Wq + (size_t)Bb * ODn * Kk;        // B*64*64  (WkxP)
constexpr size_t F_WvT = F_Wk + (size_t)Bb * ODn * Kk;       // B*64*512 (WvxP transposed [k][c])
constexpr size_t F_fat = F_WvT + (size_t)Bb * Kk * Cc;       // B*64*64
constexpr size_t F_Mm  = F_fat + (size_t)Bb * Kk * Kk;       // B*64*64  (A@fatt)
constexpr size_t F_gv  = F_Mm + (size_t)Bb * Kk * Kk;        // B*64
constexpr size_t F_Ap  = F_gv + (size_t)Bb * Kk;             // B*NCH*64*64 (A partials)
constexpr size_t F_sp  = F_Ap + (size_t)Bb * NCH * Kk * Kk;  // B*64 (sum_lamdkP), region oversized
constexpr size_t F_li  = F_sp + (size_t)Bb * NCH * Kk;       // B*HW (1/lamdv)
constexpr size_t F_Uq  = F_li + (size_t)Bb * HWn;            // B*64*HW  (P^T/lamdq)
constexpr size_t F_t   = F_Uq;                               // row-DCT temp (dead before Uq written)
constexpr size_t F_Uk  = F_Uq + (size_t)Bb * Kk * HWn;       // B*64*HW  (P^T/lamdk)
constexpr size_t F_Rs  = F_Uk;                               // Rs aliases Uk (Uk dead by then)

typedef float v2f __attribute__((ext_vector_type(2)));
typedef float v8f __attribute__((ext_vector_type(8)));

__device__ __forceinline__ v8f wmma_f32(v2f a, v2f b, v8f c) {
  // V_WMMA_F32_16X16X4_F32 : D = A(16x4) * B(4x16) + C(16x16)
  return __builtin_amdgcn_wmma_f32_16x16x4_f32(false, a, false, b, (short)0, c,
                                               false, false);
}

// ---------------- K0: DCT tables Dh, Dv and column sums of P ----------------
__global__ void k_tables(float* ws) {
  __shared__ float sDh[1024];
  __shared__ float sDv[1024];
  int h = threadIdx.x;  // 0..127
  const float PI = 3.14159265358979323846f;
  float a0  = 1.0f / sqrtf(128.0f);
  float amp = sqrtf(2.0f) / sqrtf(128.0f);
#pragma unroll
  for (int u = 0; u < 8; ++u) {
    float ang = (float)u * (2.0f * (float)h + 1.0f) * PI / 256.0f;
    float vh  = (u == 0) ? a0 : amp * cosf(ang);
    sDh[h * 8 + u] = vh;
    sDv[h * 8 + u] = vh;  // H==W and identical amplitude in the reference
    ws[F_Dh + h * 8 + u] = vh;
    ws[F_Dv + h * 8 + u] = vh;
  }
  __syncthreads();
  if (h < 64) {
    int u = h >> 3, v = h & 7;
    float su = 0.f, sv = 0.f;
    for (int i = 0; i < 128; ++i) { su += sDh[i * 8 + u]; sv += sDv[i * 8 + v]; }
    ws[F_sP + h] = su * sv;  // sum_n P[n,k] = (sum_h Dh[h,u]) * (sum_w Dv[w,v])
  }
}

// ---------------- K0b: P^T table, k-major [64][HW] ----------------
__global__ void k_pt(float* __restrict__ ws) {
  int n = blockIdx.x * 128 + threadIdx.x;
  int h = n >> 7, w = n & 127;
  float dh[8], dv[8];
#pragma unroll
  for (int i = 0; i < 8; ++i) {
    dh[i] = ws[F_Dh + (size_t)h * 8 + i];
    dv[i] = ws[F_Dv + (size_t)w * 8 + i];
  }
#pragma unroll
  for (int uu = 0; uu < 8; ++uu)
#pragma unroll
    for (int vv = 0; vv < 8; ++vv)
      ws[F_PT + (size_t)(uu * 8 + vv) * HWn + n] = dh[uu] * dv[vv];
}

// ---------------- K1: row DCT  t[b,c,h,v] = sum_w x * Dv[w,v] (one wave/row) ----------------
__global__ void k_rowdct(const float* __restrict__ x, float* __restrict__ ws) {
  size_t row = (size_t)blockIdx.x * 4 + (threadIdx.x >> 5);  // (b*C+c)*H + h
  int lane = threadIdx.x & 31;
  const float4* x4 = (const float4*)x;
  float4 xx = x4[row * 32 + lane];  // coalesced 128B per wave-quarter
  float xv[4] = {xx.x, xx.y, xx.z, xx.w};
  const float* Dv = ws + F_Dv;
  float acc[8];
#pragma unroll
  for (int v = 0; v < 8; ++v) acc[v] = 0.f;
#pragma unroll
  for (int j = 0; j < 4; ++j) {
    const float* dr = Dv + (size_t)(lane * 4 + j) * 8;
#pragma unroll
    for (int v = 0; v < 8; ++v) acc[v] += xv[j] * dr[v];
  }
#pragma unroll
  for (int v = 0; v < 8; ++v) {
    acc[v] += __shfl_xor(acc[v], 16, 32);
    acc[v] += __shfl_xor(acc[v], 8, 32);
    acc[v] += __shfl_xor(acc[v], 4, 32);
    acc[v] += __shfl_xor(acc[v], 2, 32);
    acc[v] += __shfl_xor(acc[v], 1, 32);
  }
  if (lane == 0) {
    float* t = ws + F_t + row * 8;
#pragma unroll
    for (int v = 0; v < 8; ++v) t[v] = acc[v];
  }
}

// ---------------- K2: column DCT  xP[b,c,u*8+v] = sum_h Dh[h,u] * t[...,h,v] ----------------
__global__ void k_coldct(float* __restrict__ ws) {
  int bc = blockIdx.x;
  int k = threadIdx.x, u = k >> 3, v = k & 7;
  const float* Dh = ws + F_Dh;
  const float* t  = ws + F_t + (size_t)bc * Hh * 8;
  float acc = 0.f;
  for (int h = 0; h < Hh; ++h) acc += Dh[h * 8 + u] * t[h * 8 + v];
  ws[F_xP + (size_t)bc * Kk + k] = acc;
}

// ---------------- K3: small projections  out = W @ xP (per batch) ----------------
__global__ void k_projW(const float* __restrict__ ws, const float* __restrict__ W,
                        float* __restrict__ out, int M, int trans) {
  int m = blockIdx.x, b = blockIdx.y, k = threadIdx.x;
  const float* xp = ws + F_xP + (size_t)b * Cc * Kk;
  const float* wr = W + (size_t)m * Cc;
  float acc = 0.f;
  for (int c = 0; c < Cc; ++c) acc += wr[c] * xp[(size_t)c * Kk + k];
  if (trans) out[((size_t)b * Kk + k) * M + m] = acc;       // [b][k][m]
  else       out[((size_t)b * M + m) * Kk + k] = acc;       // [b][m][k]
}

// ---------------- K4: fatt[b,k,l] = sum_o WkxP[o,k] * WqxP[o,l] ----------------
__global__ void k_fatt(float* __restrict__ ws) {
  int b = blockIdx.y;
  int p = blockIdx.x * 256 + threadIdx.x;  // (k,l) pair
  int k = p >> 6, l = p & 63;
  const float* wq = ws + F_Wq + (size_t)b * 4096;
  const float* wk = ws + F_Wk + (size_t)b * 4096;
  float acc = 0.f;
  for (int o = 0; o < 64; ++o) acc += wk[o * 64 + k] * wq[o * 64 + l];
  ws[F_fat + (size_t)b * 4096 + p] = acc;
}

// ---------------- K5 (WMMA): column norms of WqxP@P^T and WkxP@P^T; write Uq, Uk ----------------
// Per 64-column chunk: two GEMMs (M=64,K=64) sharing the B operand (P^T tile),
// then sum-of-squares over M in-register + shfl + LDS, then Uq/Uk = P^T * (1/norm).
__global__ void k_norms(float* __restrict__ ws) {
  __shared__ float pq[4][64], pk[4][64], siq[64], sik[64];
  int b = blockIdx.y;
  int n0 = blockIdx.x * 64;
  int wv = threadIdx.x >> 5, lane = threadIdx.x & 31;
  int hi = lane >> 4, l15 = lane & 15;
  int o0 = wv * 16;
  const float* wq = ws + F_Wq + (size_t)b * 4096;
  const float* wk = ws + F_Wk + (size_t)b * 4096;
  const float* PT = ws + F_PT;

  v8f z = {0.f, 0.f, 0.f, 0.f, 0.f, 0.f, 0.f, 0.f};
  v8f aq[4] = {z, z, z, z};
  v8f ak[4] = {z, z, z, z};
  for (int ks = 0; ks < 16; ++ks) {
    int ka = ks * 4 + 2 * hi;
    v2f a1 = *(const v2f*)(wq + (size_t)(o0 + l15) * 64 + ka);  // A rows are 8B pairs
    v2f a2 = *(const v2f*)(wk + (size_t)(o0 + l15) * 64 + ka);
#pragma unroll
    for (int nt = 0; nt < 4; ++nt) {
      int col = n0 + nt * 16 + l15;
      v2f bb;
      bb.x = PT[(size_t)ka * HWn + col];
      bb.y = PT[(size_t)(ka + 1) * HWn + col];
      aq[nt] = wmma_f32(a1, bb, aq[nt]);
      ak[nt] = wmma_f32(a2, bb, ak[nt]);
    }
  }
#pragma unroll
  for (int nt = 0; nt < 4; ++nt) {
    float sq = 0.f, sk2 = 0.f;
#pragma unroll
    for (int r = 0; r < 8; ++r) {
      sq  += aq[nt][r] * aq[nt][r];
      sk2 += ak[nt][r] * ak[nt][r];
    }
    sq  += __shfl_xor(sq, 16, 32);   // combine rows r and r+8 halves
    sk2 += __shfl_xor(sk2, 16, 32);
    if (lane < 16) { pq[wv][nt * 16 + l15] = sq; pk[wv][nt * 16 + l15] = sk2; }
  }
  __syncthreads();
  int t = threadIdx.x;
  if (t < 64) {
    siq[t] = 1.0f / sqrtf(pq[0][t] + pq[1][t] + pq[2][t] + pq[3][t]);
    sik[t] = 1.0f / sqrtf(pk[0][t] + pk[1][t] + pk[2][t] + pk[3][t]);
  }
  __syncthreads();
  float* Uq = ws + F_Uq + (size_t)b * Kk * HWn;
  float* Uk = ws + F_Uk + (size_t)b * Kk * HWn;
  int r = t >> 1, half = t & 1;                 // row of P^T, half of the 64 cols
  const float* pr = PT + (size_t)r * HWn + n0 + half * 32;
  float* uqr = Uq + (size_t)r * HWn + n0 + half * 32;
  float* ukr = Uk + (size_t)r * HWn + n0 + half * 32;
#pragma unroll
  for (int c = 0; c < 32; ++c) {
    float p = pr[c];
    uqr[c] = p * siq[half * 32 + c];
    ukr[c] = p * sik[half * 32 + c];
  }
}

// ---------------- K6a (WMMA): split-K partials of A = P^T x Uk^T ----------------
__global__ void k_accA(float* __restrict__ ws) {
  int ch = blockIdx.x, b = blockIdx.y;
  int wv = threadIdx.x >> 5, lane = threadIdx.x & 31;
  int hi = lane >> 4, l15 = lane & 15;
  int m0 = wv * 16;                 // A row tile (k index)
  int nbase = ch * (HWn / NCH);     // 512-wide reduction chunk
  const float* PT = ws + F_PT;
  const float* Uk = ws + F_Uk + (size_t)b * Kk * HWn;

  v8f z = {0.f, 0.f, 0.f, 0.f, 0.f, 0.f, 0.f, 0.f};
  v8f acc[4] = {z, z, z, z};
  for (int ks = 0; ks < (HWn / NCH) / 4; ++ks) {
    int ka = nbase + ks * 4 + 2 * hi;
    v2f a = *(const v2f*)(PT + (size_t)(m0 + l15) * HWn + ka);  // n-adjacent pair
#pragma unroll
    for (int nt = 0; nt < 4; ++nt) {
      v2f bb = *(const v2f*)(Uk + (size_t)(nt * 16 + l15) * HWn + ka);
      acc[nt] = wmma_f32(a, bb, acc[nt]);
    }
  }
  float* Ap = ws + F_Ap + (size_t)(b * NCH + ch) * 4096;
#pragma unroll
  for (int nt = 0; nt < 4; ++nt) {
#pragma unroll
    for (int r = 0; r < 8; ++r) {
      int m = m0 + r + hi * 8;
      Ap[(size_t)m * 64 + nt * 16 + l15] = acc[nt][r];
    }
  }
}

// ---------------- K6s: sum_lamdkP[b,k] = sum_n Uk[k,n] ----------------
__global__ void k_sumUk(float* __restrict__ ws) {
  __shared__ float red[256];
  int b = blockIdx.x, t = threadIdx.x;
  int k = t >> 2, q = t & 3;
  const float* Uk = ws + F_Uk + (size_t)b * Kk * HWn + (size_t)k * HWn + q * (HWn / 4);
  float s = 0.f;
  for (int i = 0; i < HWn / 4; ++i) s += Uk[i];
  red[t] = s;
  __syncthreads();
  if (q == 0) ws[F_sp + (size_t)b * 64 + k] = red[t] + red[t + 1] + red[t + 2] + red[t + 3];
}

// ---------------- K6b: reduce A, then Mm = A@fatt and g = sum_lamdkP@fatt ----------------
__global__ void k_reduceA(float* __restrict__ ws) {
  __shared__ float Ash[4096];
  __shared__ float ssh[64];
  int b = blockIdx.x, t = threadIdx.x;
  float s[16];
#pragma unroll
  for (int e = 0; e < 16; ++e) s[e] = 0.f;
  for (int ch = 0; ch < NCH; ++ch) {
    const float* Ap = ws + F_Ap + (size_t)(b * NCH + ch) * 4096 + (size_t)t * 16;
#pragma unroll
    for (int e = 0; e < 16; ++e) s[e] += Ap[e];
  }
#pragma unroll
  for (int e = 0; e < 16; ++e) Ash[t * 16 + e] = s[e];
  if (t < 64) ssh[t] = ws[F_sp + (size_t)b * 64 + t];
  __syncthreads();
  const float* fat = ws + F_fat + (size_t)b * 4096;
  for (int e = 0; e < 16; ++e) {
    int idx = t * 16 + e;
    int k = idx >> 6, l = idx & 63;
    float mm = 0.f;
    for (int m = 0; m < 64; ++m) mm += Ash[k * 64 + m] * fat[m * 64 + l];
    ws[F_Mm + (size_t)b * 4096 + idx] = mm;
  }
  if (t < 64) {
    float gg = 0.f;
    for (int k = 0; k < 64; ++k) gg += ssh[k] * fat[k * 64 + t];
    ws[F_gv + (size_t)b * 64 + t] = gg;
  }
}

// ---------------- K7a: 1/lamdv[b,n] = 1/(HW + g . Uq[:,n]) ----------------
__global__ void k_linv(float* __restrict__ ws) {
  int b = blockIdx.y;
  int n = blockIdx.x * 128 + threadIdx.x;
  const float* g  = ws + F_gv + (size_t)b * 64;
  const float* Uq = ws + F_Uq + (size_t)b * Kk * HWn;
  float acc = (float)HWn;
#pragma unroll
  for (int l = 0; l < 64; ++l) acc += g[l] * Uq[(size_t)l * HWn + n];
  ws[F_li + (size_t)b * HWn + n] = 1.0f / acc;
}

// ---------------- K7 (WMMA): Rs[b,k,n] = (Mm@Uq[:,n] + sP[k]) * linv[n] ----------------
__global__ void k_rs(float* __restrict__ ws) {
  int b = blockIdx.y;
  int n0 = blockIdx.x * 64;
  int wv = threadIdx.x >> 5, lane = threadIdx.x & 31;
  int hi = lane >> 4, l15 = lane & 15;
  int m0 = wv * 16;
  const float* Mm = ws + F_Mm + (size_t)b * 4096;
  const float* Uq = ws + F_Uq + (size_t)b * Kk * HWn;
  const float* sP = ws + F_sP;
  const float* li = ws + F_li + (size_t)b * HWn;
  float* Rs = ws + F_Rs + (size_t)b * Kk * HWn;

  v8f z = {0.f, 0.f, 0.f, 0.f, 0.f, 0.f, 0.f, 0.f};
  v8f acc[4] = {z, z, z, z};
  for (int ks = 0; ks < 16; ++ks) {
    int ka = ks * 4 + 2 * hi;  // this lane's K-pair (ISA A/B f32 layout)
    v2f a = *(const v2f*)(Mm + (size_t)(m0 + l15) * 64 + ka);
#pragma unroll
    for (int nt = 0; nt < 4; ++nt) {
      int col = n0 + nt * 16 + l15;
      v2f bb;
      bb.x = Uq[(size_t)ka * HWn + col];
      bb.y = Uq[(size_t)(ka + 1) * HWn + col];
      acc[nt] = wmma_f32(a, bb, acc[nt]);
    }
  }
#pragma unroll
  for (int nt = 0; nt < 4; ++nt) {
    int col = n0 + nt * 16 + l15;
    float lv = li[col];
#pragma unroll
    for (int r = 0; r < 8; ++r) {
      int m = m0 + r + hi * 8;  // C/D layout: VGPR r holds rows r (lanes<16) / r+8
      Rs[(size_t)m * HWn + col] = (acc[nt][r] + sP[m]) * lv;
    }
  }
}

// ---------------- K8 (WMMA): Out = WvxP @ Rs, fused gamma*(.+bias)+x ----------------
__global__ void k_out(const float* __restrict__ x, const float* __restrict__ bias,
                      const float* __restrict__ gam, float* __restrict__ out,
                      const float* __restrict__ ws) {
  int b = blockIdx.z;
  int n0 = blockIdx.x * 64;
  int wv = threadIdx.x >> 5, lane = threadIdx.x & 31;
  int hi = lane >> 4, l15 = lane & 15;
  int m0 = (blockIdx.y * 4 + wv) * 16;
  const float* Av = ws + F_WvT + (size_t)b * Kk * Cc;  // [k][c]
  const float* Rs = ws + F_Rs + (size_t)b * Kk * HWn;  // [k][n]

  v8f z = {0.f, 0.f, 0.f, 0.f, 0.f, 0.f, 0.f, 0.f};
  v8f acc[4] = {z, z, z, z};
  for (int ks = 0; ks < 16; ++ks) {
    int ka = ks * 4 + 2 * hi;
    v2f a;
    a.x = Av[(size_t)ka * Cc + m0 + l15];        // contiguous across lanes
    a.y = Av[(size_t)(ka + 1) * Cc + m0 + l15];
#pragma unroll
    for (int nt = 0; nt < 4; ++nt) {
      int col = n0 + nt * 16 + l15;
      v2f bb;
      bb.x = Rs[(size_t)ka * HWn + col];
      bb.y = Rs[(size_t)(ka + 1) * HWn + col];
      acc[nt] = wmma_f32(a, bb, acc[nt]);
    }
  }
  float gm = gam[0];
#pragma unroll
  for (int nt = 0; nt < 4; ++nt) {
    int col = n0 + nt * 16 + l15;
#pragma unroll
    for (int r = 0; r < 8; ++r) {
      int m = m0 + r + hi * 8;
      size_t idx = ((size_t)b * Cc + m) * HWn + col;
      out[idx] = gm * (acc[nt][r] + bias[m]) + x[idx];
    }
  }
}

// ---------------- launcher ----------------
extern "C" void kernel_launch(void* const* d_in, const int* in_sizes, int n_in,
                              void* d_out, int out_size, void* d_ws, size_t ws_size,
                              hipStream_t stream) {
  (void)in_sizes; (void)n_in; (void)out_size; (void)ws_size;
  const float* x    = (const float*)d_in[0];
  const float* Wq   = (const float*)d_in[1];
  const float* Wk   = (const float*)d_in[2];
  const float* Wv   = (const float*)d_in[3];
  const float* bias = (const float*)d_in[4];
  const float* gam  = (const float*)d_in[5];
  float* out = (float*)d_out;
  float* ws  = (float*)d_ws;

  k_tables<<<dim3(1), 128, 0, stream>>>(ws);
  k_pt<<<dim3(HWn / 128), 128, 0, stream>>>(ws);
  k_rowdct<<<dim3(Bb * Cc * Hh / 4), 128, 0, stream>>>(x, ws);
  k_coldct<<<dim3(Bb * Cc), 64, 0, stream>>>(ws);
  k_projW<<<dim3(ODn, Bb), 64, 0, stream>>>(ws, Wq, ws + F_Wq, ODn, 0);
  k_projW<<<dim3(ODn, Bb), 64, 0, stream>>>(ws, Wk, ws + F_Wk, ODn, 0);
  k_projW<<<dim3(Cc, Bb), 64, 0, stream>>>(ws, Wv, ws + F_WvT, Cc, 1);
  k_fatt<<<dim3(16, Bb), 256, 0, stream>>>(ws);
  k_norms<<<dim3(HWn / 64, Bb), 128, 0, stream>>>(ws);
  k_accA<<<dim3(NCH, Bb), 128, 0, stream>>>(ws);
  k_sumUk<<<dim3(Bb), 256, 0, stream>>>(ws);
  k_reduceA<<<dim3(Bb), 256, 0, stream>>>(ws);
  k_linv<<<dim3(HWn / 128, Bb), 128, 0, stream>>>(ws);
  k_rs<<<dim3(HWn / 64, Bb), 128, 0, stream>>>(ws);
  k_out<<<dim3(HWn / 64, Cc / 64, Bb), 128, 0, stream>>>(x, bias, gam, out, ws);
}